// SampleAndAggregate_87325275062516
// MI455X (gfx1250) — compile-verified
//
#include <hip/hip_runtime.h>
#include <hip/hip_bf16.h>

typedef __attribute__((ext_vector_type(2))) float v2f;
typedef __attribute__((ext_vector_type(8))) float v8f;

// -------- hop features: dst[i,0:128]=feat[s[i]], [128:160]=nt(a), [160:192]=nt(s)+nt(a)
__global__ __launch_bounds__(192) void hop_feat(
    const float* __restrict__ feat, const float* __restrict__ nt_emb,
    const int* __restrict__ ntype, const int* __restrict__ s,
    const int* __restrict__ a, float* __restrict__ dst)
{
    const int i = blockIdx.x;
    const int t = threadIdx.x;
    const int si = s[i];
    const int ai = a[i];
    float v;
    if (t < 128) {
        v = feat[(size_t)si * 128 + t];
    } else if (t < 160) {
        v = nt_emb[ntype[ai] * 32 + (t - 128)];
    } else {
        v = nt_emb[ntype[si] * 32 + (t - 160)] + nt_emb[ntype[ai] * 32 + (t - 160)];
    }
    dst[(size_t)i * 192 + t] = v;
}

// -------- fused hop-2 gather + mean over 25 neighbors: dst [nrows,192]
__global__ __launch_bounds__(192) void hop_feat_mean25(
    const float* __restrict__ feat, const float* __restrict__ nt_emb,
    const int* __restrict__ ntype, const int* __restrict__ s,
    const int* __restrict__ a, float* __restrict__ dst)
{
    const int i = blockIdx.x;
    const int t = threadIdx.x;
    __shared__ int ss[25];   // sample node index
    __shared__ int ts[25];   // type of sample node
    __shared__ int ta[25];   // type of adj node
    if (t < 25) {
        int si = s[i * 25 + t];
        ss[t] = si;
        ts[t] = ntype[si];
    }
    if (t >= 32 && t < 57) {
        int k = t - 32;
        ta[k] = ntype[a[i * 25 + k]];
    }
    __syncthreads();

    float acc = 0.0f;
    if (t < 128) {
        #pragma unroll 5
        for (int k = 0; k < 25; ++k)
            acc += feat[(size_t)ss[k] * 128 + t];
    } else if (t < 160) {
        const int c = t - 128;
        #pragma unroll 5
        for (int k = 0; k < 25; ++k)
            acc += nt_emb[ta[k] * 32 + c];
    } else {
        const int c = t - 160;
        #pragma unroll 5
        for (int k = 0; k < 25; ++k)
            acc += nt_emb[ts[k] * 32 + c] + nt_emb[ta[k] * 32 + c];
    }
    dst[(size_t)i * 192 + t] = acc / 25.0f;
}

// -------- mean over `fan` consecutive rows of width `dim`
__global__ void mean_rows(const float* __restrict__ src, float* __restrict__ dst,
                          int fan, int dim)
{
    const int i = blockIdx.x;
    const float inv = 1.0f / (float)fan;
    for (int t = threadIdx.x; t < dim; t += blockDim.x) {
        float acc = 0.0f;
        for (int k = 0; k < fan; ++k)
            acc += src[((size_t)i * fan + k) * dim + t];
        dst[(size_t)i * dim + t] = acc * inv;
    }
}

// -------- f32 WMMA GEMM: C[16*gridX, N] (ldc) = act(A[M,K] @ W[K,N])
// one wave -> one 16x16 tile; block = 8 waves -> 16 rows x 128 cols
__global__ __launch_bounds__(256) void wmma_gemm_f32(
    const float* __restrict__ A, const float* __restrict__ W,
    float* __restrict__ C, int K, int N, int ldc, int relu)
{
    const int wave = threadIdx.x >> 5;
    const int lane = threadIdx.x & 31;
    const int row0 = blockIdx.x * 16;
    const int col0 = blockIdx.y * 128 + wave * 16;

    const int l = lane & 15;   // M index (A) / N index (B, C)
    const int h = lane >> 4;   // lane-half selects K pair (A,B) / M+8 (C)

    // A 16x4 layout: lanes 0-15 hold K = {k, k+1}, lanes 16-31 hold K = {k+2, k+3}
    const float* Ap = A + (size_t)(row0 + l) * K + 2 * h;
    // B 4x16 layout mirrored: row (k+2h) in .x, row (k+2h+1) in .y, col = col0+l
    const float* Wp = W + (size_t)(2 * h) * N + col0 + l;

    v8f c = {};
    #pragma unroll 4
    for (int k = 0; k < K; k += 4) {
        v2f av = *(const v2f*)(Ap + k);          // 8B-aligned: K,k even
        v2f bv;
        bv.x = Wp[(size_t)k * N];
        bv.y = Wp[(size_t)(k + 1) * N];
        c = __builtin_amdgcn_wmma_f32_16x16x4_f32(
            false, av, false, bv, (short)0, c, false, false);
    }

    // C layout: VGPR r, lanes 0-15 -> M = r, lanes 16-31 -> M = r + 8
    float* Cp = C + (size_t)(row0 + 8 * h) * ldc + col0 + l;
    #pragma unroll
    for (int r = 0; r < 8; ++r) {
        float v = c[r];
        if (relu) v = fmaxf(v, 0.0f);
        Cp[(size_t)r * ldc] = v;
    }
}

extern "C" void kernel_launch(void* const* d_in, const int* in_sizes, int n_in,
                              void* d_out, int out_size, void* d_ws, size_t ws_size,
                              hipStream_t stream) {
    const float* features = (const float*)d_in[0];
    const float* nt_emb   = (const float*)d_in[1];
    const float* w_self0  = (const float*)d_in[2];
    const float* w_neigh0 = (const float*)d_in[3];
    const float* w_self1  = (const float*)d_in[4];
    const float* w_neigh1 = (const float*)d_in[5];
    const int*   ntype    = (const int*)d_in[6];
    const int*   s0 = (const int*)d_in[7];
    const int*   s1 = (const int*)d_in[8];
    const int*   s2 = (const int*)d_in[9];
    const int*   a0 = (const int*)d_in[10];
    const int*   a1 = (const int*)d_in[11];
    const int*   a2 = (const int*)d_in[12];
    float* out = (float*)d_out;

    // workspace layout (floats)
    float* ws      = (float*)d_ws;
    float* hidden1 = ws;                                   // 10240*192
    float* h2mean  = hidden1 + (size_t)10240 * 192;        // 10240*192
    float* hidden0 = h2mean  + (size_t)10240 * 192;        // 1024*192
    float* h1mean  = hidden0 + (size_t)1024 * 192;         // 1024*192
    float* out1    = h1mean  + (size_t)1024 * 192;         // 10240*512
    float* out0    = out1    + (size_t)10240 * 512;        // 1024*512
    float* out1m   = out0    + (size_t)1024 * 512;         // 1024*512

    // hop features
    hop_feat<<<1024,  192, 0, stream>>>(features, nt_emb, ntype, s0, a0, hidden0);
    hop_feat<<<10240, 192, 0, stream>>>(features, nt_emb, ntype, s1, a1, hidden1);
    hop_feat_mean25<<<10240, 192, 0, stream>>>(features, nt_emb, ntype, s2, a2, h2mean);
    mean_rows<<<1024, 192, 0, stream>>>(hidden1, h1mean, 10, 192);

    // layer 0: out = relu([self @ w_self0 , neigh_mean @ w_neigh0])  (ldc = 512)
    wmma_gemm_f32<<<dim3(640, 2), 256, 0, stream>>>(hidden1, w_self0,  out1,       192, 256, 512, 1);
    wmma_gemm_f32<<<dim3(640, 2), 256, 0, stream>>>(h2mean,  w_neigh0, out1 + 256, 192, 256, 512, 1);
    wmma_gemm_f32<<<dim3(64,  2), 256, 0, stream>>>(hidden0, w_self0,  out0,       192, 256, 512, 1);
    wmma_gemm_f32<<<dim3(64,  2), 256, 0, stream>>>(h1mean,  w_neigh0, out0 + 256, 192, 256, 512, 1);

    // layer 1: mean over fanout-10, then identity-activation GEMMs into d_out
    mean_rows<<<1024, 256, 0, stream>>>(out1, out1m, 10, 512);
    wmma_gemm_f32<<<dim3(64, 2), 256, 0, stream>>>(out0,  w_self1,  out,       512, 256, 512, 0);
    wmma_gemm_f32<<<dim3(64, 2), 256, 0, stream>>>(out1m, w_neigh1, out + 256, 512, 256, 512, 0);
}